// VSSBlock_61400852464203
// MI455X (gfx1250) — compile-verified
//
#include <hip/hip_runtime.h>
#include <math.h>

// ---------------- constants ----------------
#define Bq   2
#define Hq   64
#define Wq   64
#define Cq   96
#define DEq  192
#define DIq  192
#define Nq   16
#define Rq   6
#define Kq   4
#define Lq   4096          // H*W
#define Pq   8192          // B*H*W
#define C48  48            // padded (R + 2N = 38) -> 48

typedef float v2f __attribute__((ext_vector_type(2)));
typedef float v8f __attribute__((ext_vector_type(8)));
typedef unsigned int v4u  __attribute__((ext_vector_type(4)));
typedef int          v8i_ __attribute__((ext_vector_type(8)));
typedef int          v4i_ __attribute__((ext_vector_type(4)));

#if defined(__has_builtin)
#if __has_builtin(__builtin_amdgcn_tensor_load_to_lds)
#define HAS_TDM 1
#endif
#endif
#ifndef HAS_TDM
#define HAS_TDM 0
#endif

__device__ __forceinline__ v8f wmma_f32_4(v2f a, v2f b, v8f c) {
  // D = A(16x4) * B(4x16) + C, fp32, wave32
  return __builtin_amdgcn_wmma_f32_16x16x4_f32(false, a, false, b, (short)0, c,
                                               false, false);
}

__device__ __forceinline__ float silu_f(float v) {
  return v / (1.f + __expf(-v));
}

#if HAS_TDM
// ---- Tensor Data Mover descriptor helpers (CDNA5 D#, ISA §8.3/§8.4) ----
__device__ __forceinline__ unsigned lds_off32(const void* p) {
  // LDS aperture: LDS offset is the low 32 bits of the flat address
  return (unsigned)(unsigned long long)(uintptr_t)p;
}

__device__ __forceinline__ v4u tdm_g0(unsigned lds, unsigned long long ga) {
  v4u g;
  g.x = 1u;                                       // count=1, user mode
  g.y = lds;                                      // lds_addr [63:32]
  g.z = (unsigned)ga;                             // global_addr[31:0]
  g.w = (unsigned)((ga >> 32) & 0x01FFFFFFull)    // global_addr[56:32]
        | (2u << 30);                             // type=2 ("image")
  return g;
}

__device__ __forceinline__ v8i_ tdm_g1(unsigned dim0, unsigned dim1,
                                       unsigned t0, unsigned t1,
                                       unsigned long long stride0) {
  v8i_ g;
  g[0] = (int)(2u << 16);                                            // data_size=4B
  g[1] = (int)((dim0 & 0xFFFFu) << 16);                              // tensor_dim0 lo
  g[2] = (int)(((dim0 >> 16) & 0xFFFFu) | ((dim1 & 0xFFFFu) << 16)); // dim0 hi|dim1 lo
  g[3] = (int)(((dim1 >> 16) & 0xFFFFu) | ((t0 & 0xFFFFu) << 16));   // dim1 hi|tile0
  g[4] = (int)(t1 & 0xFFFFu);                                        // tile1, tile2=0
  g[5] = (int)(unsigned)(stride0 & 0xFFFFFFFFull);                   // stride0 lo32
  g[6] = (int)(unsigned)((stride0 >> 32) & 0xFFFFull);               // stride0 hi16
  g[7] = 0;
  return g;
}

__device__ __forceinline__ void tdm_load(v4u g0, v8i_ g1) {
  v4i_ z4 = {0, 0, 0, 0};
#if __clang_major__ >= 23
  v8i_ z8 = {0, 0, 0, 0, 0, 0, 0, 0};
  __builtin_amdgcn_tensor_load_to_lds(g0, g1, z4, z4, z8, 0);
#else
  __builtin_amdgcn_tensor_load_to_lds(g0, g1, z4, z4, 0);
#endif
}
#endif // HAS_TDM

// ---------------- 1) pre-LayerNorm over C=96, one wave per pixel ----------
__global__ void k_ln_pre(const float* __restrict__ x, const float* __restrict__ g,
                         const float* __restrict__ bb, float* __restrict__ xn) {
  int p = blockIdx.x * 8 + (threadIdx.x >> 5);
  int lane = threadIdx.x & 31;
  if (p >= Pq) return;
  const float* row = x + (size_t)p * Cq;
  float v0 = row[lane], v1 = row[lane + 32], v2 = row[lane + 64];
  float s = v0 + v1 + v2;
  for (int o = 16; o; o >>= 1) s += __shfl_xor(s, o, 32);
  float mu = s * (1.f / 96.f);
  float d0 = v0 - mu, d1 = v1 - mu, d2 = v2 - mu;
  float q = d0 * d0 + d1 * d1 + d2 * d2;
  for (int o = 16; o; o >>= 1) q += __shfl_xor(q, o, 32);
  float inv = rsqrtf(q * (1.f / 96.f) + 1e-6f);
  float* orow = xn + (size_t)p * Cq;
  orow[lane]      = d0 * inv * g[lane]      + bb[lane];
  orow[lane + 32] = d1 * inv * g[lane + 32] + bb[lane + 32];
  orow[lane + 64] = d2 * inv * g[lane + 64] + bb[lane + 64];
}

// ---------------- 2) in_proj GEMM: [8192x96]x[96x384], WMMA f32 ----------
__global__ void k_gemm_inproj(const float* __restrict__ xn,
                              const float* __restrict__ W,
                              float* __restrict__ xh, float* __restrict__ zs) {
  const int NT = 24;                                  // 384/16
  int tile = blockIdx.x * 4 + (threadIdx.x >> 5);     // wave id -> tile
  int mt = tile / NT, nt = tile % NT;
  int lane = threadIdx.x & 31;
  int fr = lane & 15;            // M row (A) / N col (B)
  int kk = (lane >> 4) * 2;      // K pair base
  int m0 = mt * 16, n0 = nt * 16;
  v8f acc = {0.f, 0.f, 0.f, 0.f, 0.f, 0.f, 0.f, 0.f};
  const float* arow = xn + (size_t)(m0 + fr) * Cq;
  const float* brow = W + (size_t)(n0 + fr) * Cq;
  for (int kb = 0; kb < Cq; kb += 4) {
    v2f a, b;
    a.x = arow[kb + kk];     a.y = arow[kb + kk + 1];
    b.x = brow[kb + kk];     b.y = brow[kb + kk + 1];
    acc = wmma_f32_4(a, b, acc);
  }
  int nn = n0 + fr;
  int mb = m0 + ((lane >> 4) * 8);
#pragma unroll
  for (int r = 0; r < 8; r++) {
    float v = acc[r];
    int m = mb + r;
    if (nn < DEq) xh[(size_t)m * DEq + nn] = v;
    else          zs[(size_t)m * DEq + (nn - DEq)] = silu_f(v);
  }
}

// ---------------- 3) depthwise 3x3 conv + SiLU, scatter to 4 directions ---
__global__ void k_conv_dw(const float* __restrict__ xh,
                          const float* __restrict__ cw,
                          const float* __restrict__ cb,
                          float* __restrict__ xs) {
  int p = blockIdx.x;            // b*4096 + h*64 + w
  int d = threadIdx.x;           // 0..191
  int b = p >> 12, l = p & (Lq - 1);
  int h = l >> 6, w = l & 63;
  float s = cb[d];
#pragma unroll
  for (int dy = -1; dy <= 1; dy++)
#pragma unroll
    for (int dx = -1; dx <= 1; dx++) {
      int hh = h + dy, ww = w + dx;
      if (hh >= 0 && hh < Hq && ww >= 0 && ww < Wq)
        s += xh[(size_t)((b << 12) + (hh << 6) + ww) * DEq + d] *
             cw[d * 9 + (dy + 1) * 3 + (dx + 1)];
    }
  s = silu_f(s);
  int l0 = (h << 6) + w;         // row-major index
  int l1 = (w << 6) + h;         // col-major traversal index
  size_t base = (size_t)b * Kq * Lq * DIq;
  xs[base + ((size_t)0 * Lq + l0) * DIq + d] = s;
  xs[base + ((size_t)1 * Lq + l1) * DIq + d] = s;
  xs[base + ((size_t)2 * Lq + (Lq - 1 - l0)) * DIq + d] = s;
  xs[base + ((size_t)3 * Lq + (Lq - 1 - l1)) * DIq + d] = s;
}

// ---------------- 4) x_proj GEMM per (b,k): [4096x192]x[192x38->48] ------
__global__ void k_gemm_xdbl(const float* __restrict__ xs,
                            const float* __restrict__ xpw,
                            float* __restrict__ dbl) {
  const int MT = 256, NT = 3;                          // 4096/16, 48/16
  int tile = blockIdx.x * 4 + (threadIdx.x >> 5);
  int bk = tile / (MT * NT);
  int t = tile % (MT * NT);
  int mt = t / NT, nt = t % NT;
  int k = bk & 3;
  int lane = threadIdx.x & 31;
  int fr = lane & 15;
  int kk = (lane >> 4) * 2;
  int m0 = mt * 16, n0 = nt * 16;
  const float* A = xs + (size_t)bk * Lq * DIq + (size_t)(m0 + fr) * DIq;
  const float* Wk = xpw + (size_t)k * 38 * DIq;
  int c = n0 + fr;
  bool cv = (c < 38);
  const float* brow = Wk + (size_t)(cv ? c : 0) * DIq;
  v8f acc = {0.f, 0.f, 0.f, 0.f, 0.f, 0.f, 0.f, 0.f};
  for (int kb = 0; kb < DIq; kb += 4) {
    v2f a, b;
    a.x = A[kb + kk];  a.y = A[kb + kk + 1];
    b.x = cv ? brow[kb + kk] : 0.f;
    b.y = cv ? brow[kb + kk + 1] : 0.f;
    acc = wmma_f32_4(a, b, acc);
  }
  int nn = n0 + fr;
  int mb = m0 + ((lane >> 4) * 8);
  float* out = dbl + (size_t)bk * Lq * C48;
#pragma unroll
  for (int r = 0; r < 8; r++)
    out[(size_t)(mb + r) * C48 + nn] = acc[r];
}

// ---------------- 5) selective scan, TDM double-buffered staging ----------
// grid (8 bk, 6 d-chunks) x 32 threads (one wave per block).
// Per 16-step chunk: TDM#1 copies 16x48 x_dbl rows (dts/B/C), TDM#2 copies a
// 32-wide x 16-row strided tile of xs. Double-buffered on TENSORcnt.
__global__ void k_scan(const float* __restrict__ xs, const float* __restrict__ dbl,
                       const float* __restrict__ dtw, const float* __restrict__ dtb,
                       const float* __restrict__ Alogs, const float* __restrict__ Ds,
                       float* __restrict__ yout) {
  int bk = blockIdx.x;                    // 0..7 = b*4+k
  int k = bk & 3;
  int tid = threadIdx.x;                  // 0..31
  int d = blockIdx.y * 32 + tid;          // 0..191
  const float* xsp  = xs  + (size_t)bk * Lq * DIq;
  const float* dblp = dbl + (size_t)bk * Lq * C48;
  float* yp = yout + (size_t)bk * Lq * DIq;
  int gd = k * DIq + d;
  float A[Nq], hst[Nq];
#pragma unroll
  for (int n = 0; n < Nq; n++) {
    A[n] = -__expf(Alogs[(size_t)gd * Nq + n]);
    hst[n] = 0.f;
  }
  float w6[Rq];
#pragma unroll
  for (int r = 0; r < Rq; r++) w6[r] = dtw[(size_t)gd * Rq + r];
  float bias = dtb[gd];
  float Dd = Ds[gd];

  __shared__ __align__(16) float smA[2][16 * C48];   // dts/B/C rows
  __shared__ __align__(16) float smX[2][16 * 32];    // xs tile for this d-range
  const int NC = Lq / 16;                            // 256 chunks

#if HAS_TDM
  const v8i_ g1A = tdm_g1(16 * C48, 1, 16 * C48, 1, 16 * C48); // 1D 768 floats
  const v8i_ g1X = tdm_g1(32, 16, 32, 16, DIq);                // 2D 32x16, stride 192
  const unsigned la[2] = {lds_off32(&smA[0][0]), lds_off32(&smA[1][0])};
  const unsigned lx[2] = {lds_off32(&smX[0][0]), lds_off32(&smX[1][0])};
  const float* xbase = xsp + (size_t)blockIdx.y * 32;

  // prologue: chunk 0 in flight
  tdm_load(tdm_g0(la[0], (unsigned long long)(uintptr_t)(dblp)), g1A);
  tdm_load(tdm_g0(lx[0], (unsigned long long)(uintptr_t)(xbase)), g1X);

  for (int c = 0; c < NC; c++) {
    int buf = c & 1;
    if (c + 1 < NC) {
      tdm_load(tdm_g0(la[buf ^ 1],
               (unsigned long long)(uintptr_t)(dblp + (size_t)(c + 1) * 16 * C48)), g1A);
      tdm_load(tdm_g0(lx[buf ^ 1],
               (unsigned long long)(uintptr_t)(xbase + (size_t)(c + 1) * 16 * DIq)), g1X);
      __builtin_amdgcn_s_wait_tensorcnt(2);   // chunk c landed, c+1 in flight
    } else {
      __builtin_amdgcn_s_wait_tensorcnt(0);
    }
    const float* A_ = &smA[buf][0];
    const float* X_ = &smX[buf][0];
    for (int s = 0; s < 16; s++) {
      float dt = bias;
#pragma unroll
      for (int r = 0; r < Rq; r++) dt += A_[s * C48 + r] * w6[r];
      dt = (dt > 20.f) ? dt : log1pf(__expf(dt));   // softplus
      float xv = X_[s * 32 + tid];
      float dx = dt * xv;
      float y = 0.f;
#pragma unroll
      for (int n = 0; n < Nq; n++) {
        hst[n] = __expf(dt * A[n]) * hst[n] + dx * A_[s * C48 + 6 + n];
        y += hst[n] * A_[s * C48 + 22 + n];
      }
      yp[(size_t)(c * 16 + s) * DIq + d] = y + xv * Dd;  // + xs*Ds
    }
  }
#else
  // fallback: manual LDS staging (single wave per block -> barriers are cheap)
  for (int c = 0; c < NC; c++) {
    int tc = c * 16;
    __syncthreads();
    for (int idx = tid; idx < 16 * C48; idx += 32)
      smA[0][idx] = dblp[(size_t)tc * C48 + idx];
    for (int idx = tid; idx < 16 * 32; idx += 32) {
      int s = idx >> 5, j = idx & 31;
      smX[0][idx] = xsp[(size_t)(tc + s) * DIq + blockIdx.y * 32 + j];
    }
    if (tc + 16 < Lq)
      __builtin_prefetch(&dblp[(size_t)(tc + 16) * C48], 0, 1);
    __syncthreads();
    for (int s = 0; s < 16; s++) {
      float dt = bias;
#pragma unroll
      for (int r = 0; r < Rq; r++) dt += smA[0][s * C48 + r] * w6[r];
      dt = (dt > 20.f) ? dt : log1pf(__expf(dt));
      float xv = smX[0][s * 32 + tid];
      float dx = dt * xv;
      float y = 0.f;
#pragma unroll
      for (int n = 0; n < Nq; n++) {
        hst[n] = __expf(dt * A[n]) * hst[n] + dx * smA[0][s * C48 + 6 + n];
        y += hst[n] * smA[0][s * C48 + 22 + n];
      }
      yp[(size_t)(tc + s) * DIq + d] = y + xv * Dd;
    }
  }
#endif
}

// ---------------- 6) direction merge + out-LayerNorm + gate ---------------
__global__ void k_combine_ln(const float* __restrict__ yout,
                             const float* __restrict__ zs,
                             const float* __restrict__ g,
                             const float* __restrict__ bb,
                             float* __restrict__ yz) {
  int p = blockIdx.x;            // b*4096 + l (row-major position)
  int d = threadIdx.x;           // 0..191
  int b = p >> 12, l = p & (Lq - 1);
  int h = l >> 6, w = l & 63;
  int lt = (w << 6) + h;         // position in col-major traversal
  size_t base = (size_t)b * Kq * Lq * DIq;
  float v = yout[base + ((size_t)0 * Lq + l) * DIq + d]
          + yout[base + ((size_t)2 * Lq + (Lq - 1 - l)) * DIq + d]
          + yout[base + ((size_t)1 * Lq + lt) * DIq + d]
          + yout[base + ((size_t)3 * Lq + (Lq - 1 - lt)) * DIq + d];
  __shared__ float red[12];
  float s = v;
  for (int o = 16; o; o >>= 1) s += __shfl_xor(s, o, 32);
  int wid = d >> 5;
  if ((d & 31) == 0) red[wid] = s;
  __syncthreads();
  float mu = (red[0] + red[1] + red[2] + red[3] + red[4] + red[5]) * (1.f / 192.f);
  float dv = v - mu;
  float q = dv * dv;
  for (int o = 16; o; o >>= 1) q += __shfl_xor(q, o, 32);
  if ((d & 31) == 0) red[6 + wid] = q;
  __syncthreads();
  float var = (red[6] + red[7] + red[8] + red[9] + red[10] + red[11]) * (1.f / 192.f);
  float ln = dv * rsqrtf(var + 1e-6f) * g[d] + bb[d];
  yz[(size_t)p * DIq + d] = ln * zs[(size_t)p * DIq + d];
}

// ---------------- 7) out_proj GEMM + residual: [8192x192]x[192x96] --------
__global__ void k_gemm_outproj(const float* __restrict__ yz,
                               const float* __restrict__ opw,
                               const float* __restrict__ xin,
                               float* __restrict__ out) {
  const int NT = 6;                                   // 96/16
  int tile = blockIdx.x * 4 + (threadIdx.x >> 5);
  int mt = tile / NT, nt = tile % NT;
  int lane = threadIdx.x & 31;
  int fr = lane & 15;
  int kk = (lane >> 4) * 2;
  int m0 = mt * 16, n0 = nt * 16;
  const float* arow = yz + (size_t)(m0 + fr) * DEq;
  const float* brow = opw + (size_t)(n0 + fr) * DEq;  // opw[c][e]
  v8f acc = {0.f, 0.f, 0.f, 0.f, 0.f, 0.f, 0.f, 0.f};
  for (int kb = 0; kb < DEq; kb += 4) {
    v2f a, b;
    a.x = arow[kb + kk];  a.y = arow[kb + kk + 1];
    b.x = brow[kb + kk];  b.y = brow[kb + kk + 1];
    acc = wmma_f32_4(a, b, acc);
  }
  int nn = n0 + fr;
  int mb = m0 + ((lane >> 4) * 8);
#pragma unroll
  for (int r = 0; r < 8; r++) {
    int m = mb + r;
    out[(size_t)m * Cq + nn] = xin[(size_t)m * Cq + nn] + acc[r];
  }
}

// ---------------- launcher ----------------
extern "C" void kernel_launch(void* const* d_in, const int* in_sizes, int n_in,
                              void* d_out, int out_size, void* d_ws, size_t ws_size,
                              hipStream_t stream) {
  (void)in_sizes; (void)n_in; (void)out_size; (void)ws_size;
  const float* x        = (const float*)d_in[0];
  const float* norm_g   = (const float*)d_in[1];
  const float* norm_b   = (const float*)d_in[2];
  const float* in_projw = (const float*)d_in[3];
  const float* conv_w   = (const float*)d_in[4];
  const float* conv_b   = (const float*)d_in[5];
  const float* x_projw  = (const float*)d_in[6];
  const float* dt_w     = (const float*)d_in[7];
  const float* dt_b     = (const float*)d_in[8];
  const float* A_logs   = (const float*)d_in[9];
  const float* Ds       = (const float*)d_in[10];
  const float* onorm_g  = (const float*)d_in[11];
  const float* onorm_b  = (const float*)d_in[12];
  const float* out_projw= (const float*)d_in[13];
  float* out = (float*)d_out;

  float* ws  = (float*)d_ws;
  float* xn  = ws;                         // 8192*96      =  786432
  float* xh  = xn  + (size_t)Pq * Cq;      // 8192*192     = 1572864
  float* zs  = xh  + (size_t)Pq * DEq;     // 8192*192
  float* xsb = zs  + (size_t)Pq * DEq;     // 8*4096*192   = 6291456
  float* dbl = xsb + (size_t)Bq * Kq * Lq * DIq;   // 8*4096*48 = 1572864
  float* yo  = dbl + (size_t)Bq * Kq * Lq * C48;   // 8*4096*192
  float* yz  = xh;                         // alias: xh dead after conv

  k_ln_pre      <<<Pq / 8, 256, 0, stream>>>(x, norm_g, norm_b, xn);
  k_gemm_inproj <<<(512 * 24) / 4, 128, 0, stream>>>(xn, in_projw, xh, zs);
  k_conv_dw     <<<Pq, DEq, 0, stream>>>(xh, conv_w, conv_b, xsb);
  k_gemm_xdbl   <<<(8 * 256 * 3) / 4, 128, 0, stream>>>(xsb, x_projw, dbl);
  k_scan        <<<dim3(8, 6), 32, 0, stream>>>(xsb, dbl, dt_w, dt_b, A_logs, Ds, yo);
  k_combine_ln  <<<Pq, DIq, 0, stream>>>(yo, zs, onorm_g, onorm_b, yz);
  k_gemm_outproj<<<(512 * 6) / 4, 128, 0, stream>>>(yz, out_projw, x, out);
}